// BaselineGCN_45019847197427
// MI455X (gfx1250) — compile-verified
//
#include <hip/hip_runtime.h>
#include <math.h>

// ---------------------------------------------------------------------------
// 2-layer GCN for MI455X (gfx1250, wave32).
//   dinv  = rsqrt(1 + indeg)                      (self-loop folded in)
//   h1    = x @ W1                                 (WMMA f32 16x16x4)
//   o1    = h1*dinv^2 + b1  then  o1[d] += h1[s]*dinv[s]*dinv[d]  (atomics)
//   h2    = relu(o1) @ W2                          (WMMA, relu fused on load)
//   o2    = h2*dinv^2 + b2  then  o2[d] += h2[s]*dinv[s]*dinv[d]
//   y     = relu(o2) @ Wo + bo
// ---------------------------------------------------------------------------

typedef float v2f __attribute__((ext_vector_type(2)));
typedef float v8f __attribute__((ext_vector_type(8)));

#define WMMA4(a, b, c) \
  __builtin_amdgcn_wmma_f32_16x16x4_f32(false, (a), false, (b), (short)0, (c), false, false)

// ---- degree / normalization ------------------------------------------------
__global__ void k_fill_one(float* __restrict__ deg, int n) {
  int i = blockIdx.x * blockDim.x + threadIdx.x;
  if (i < n) deg[i] = 1.0f;  // self-loop contribution
}

__global__ void k_deg_accum(const int* __restrict__ dst, float* __restrict__ deg, int E) {
  int i = blockIdx.x * blockDim.x + threadIdx.x;
  int stride = gridDim.x * blockDim.x;
  for (; i < E; i += stride) atomicAdd(&deg[dst[i]], 1.0f);
}

__global__ void k_rsqrt_inplace(float* __restrict__ deg, int n) {
  int i = blockIdx.x * blockDim.x + threadIdx.x;
  if (i < n) deg[i] = rsqrtf(deg[i]);  // deg >= 1 always (self loops)
}

// ---- GEMM1: h1[N,32] = x[N,128] @ W1[128,32]  (one wave -> 16x32 tile) -----
__global__ void k_gemm1(const float* __restrict__ x, const float* __restrict__ W1,
                        float* __restrict__ h1, int n) {
  const int wave = (blockIdx.x * blockDim.x + threadIdx.x) >> 5;
  const int lane = threadIdx.x & 31;
  const int m0 = wave * 16;
  if (m0 >= n) return;  // wave-uniform: EXEC stays all-ones for WMMA
  const int half = lane >> 4;  // 0: K pair {kb,kb+1}; 1: {kb+2,kb+3} (ISA A layout)
  const int l16 = lane & 15;
  const int row = min(m0 + l16, n - 1);  // clamp loads; tail stores guarded below

  v8f acc0 = {};
  v8f acc1 = {};
  const float* arow = x + (size_t)row * 128 + half * 2;
#pragma unroll 4
  for (int kb = 0; kb < 128; kb += 4) {
    v2f a;
    a.x = arow[kb + 0];
    a.y = arow[kb + 1];
    const float* brow = W1 + (size_t)(kb + half * 2) * 32 + l16;
    v2f b0, b1;
    b0.x = brow[0];
    b0.y = brow[32];
    b1.x = brow[16];
    b1.y = brow[48];
    acc0 = WMMA4(a, b0, acc0);
    acc1 = WMMA4(a, b1, acc1);
  }
  // C/D layout: VGPR j -> row m0+j (lanes 0-15) / m0+j+8 (lanes 16-31), col l16
  if (m0 + 16 <= n) {  // wave-uniform fast path: no per-row guards
    float* p = h1 + (size_t)(m0 + half * 8) * 32 + l16;
#pragma unroll
    for (int j = 0; j < 8; ++j) {
      p[(size_t)j * 32] = acc0[j];
      p[(size_t)j * 32 + 16] = acc1[j];
    }
  } else {  // ragged tail
#pragma unroll
    for (int j = 0; j < 8; ++j) {
      int m = m0 + j + half * 8;
      if (m < n) {
        h1[(size_t)m * 32 + l16] = acc0[j];
        h1[(size_t)m * 32 + 16 + l16] = acc1[j];
      }
    }
  }
}

// ---- GEMM2: h2[N,16] = relu(o1)[N,32] @ W2[32,16] --------------------------
__global__ void k_gemm2(const float* __restrict__ o1, const float* __restrict__ W2,
                        float* __restrict__ h2, int n) {
  const int wave = (blockIdx.x * blockDim.x + threadIdx.x) >> 5;
  const int lane = threadIdx.x & 31;
  const int m0 = wave * 16;
  if (m0 >= n) return;
  const int half = lane >> 4;
  const int l16 = lane & 15;
  const int row = min(m0 + l16, n - 1);

  v8f acc = {};
  const float* arow = o1 + (size_t)row * 32 + half * 2;
#pragma unroll
  for (int kb = 0; kb < 32; kb += 4) {
    v2f a;
    a.x = fmaxf(arow[kb + 0], 0.0f);  // fused ReLU of layer-1 output
    a.y = fmaxf(arow[kb + 1], 0.0f);
    const float* brow = W2 + (size_t)(kb + half * 2) * 16 + l16;
    v2f b;
    b.x = brow[0];
    b.y = brow[16];
    acc = WMMA4(a, b, acc);
  }
  if (m0 + 16 <= n) {
    float* p = h2 + (size_t)(m0 + half * 8) * 16 + l16;
#pragma unroll
    for (int j = 0; j < 8; ++j) p[(size_t)j * 16] = acc[j];
  } else {
#pragma unroll
    for (int j = 0; j < 8; ++j) {
      int m = m0 + j + half * 8;
      if (m < n) h2[(size_t)m * 16 + l16] = acc[j];
    }
  }
}

// ---- self-loop + bias init: out[i,f] = h[i,f]*dinv[i]^2 + bias[f] ----------
// float4-vectorized; F is 16 or 32 so a float4 never crosses a row boundary.
__global__ void k_selfinit(const float4* __restrict__ h4, const float* __restrict__ dinv,
                           const float4* __restrict__ bias4, float4* __restrict__ out4,
                           int n, int logF) {
  const int logF4 = logF - 2;
  int idx = blockIdx.x * blockDim.x + threadIdx.x;
  if (idx >= (n << logF4)) return;
  int i = idx >> logF4;
  int f4 = idx & ((1 << logF4) - 1);
  float dv = dinv[i];
  float s = dv * dv;
  float4 hv = h4[idx];
  float4 bv = bias4[f4];
  float4 o;
  o.x = hv.x * s + bv.x;
  o.y = hv.y * s + bv.y;
  o.z = hv.z * s + bv.z;
  o.w = hv.w * s + bv.w;
  out4[idx] = o;
}

// ---- edge aggregation: out[d,f] += h[s,f] * dinv[s]*dinv[d] ----------------
// idx = e*F + f : the F lanes of an edge are consecutive -> coalesced gather
// of h[s,*] and coalesced same-cacheline atomics on out[d,*] (L2-resident;
// h and out tables are <= 12.8 MB vs 192 MB L2).
__global__ void k_agg(const int* __restrict__ src, const int* __restrict__ dst,
                      const float* __restrict__ dinv, const float* __restrict__ h,
                      float* __restrict__ out, int E, int logF) {
  const int F = 1 << logF;
  int idx = blockIdx.x * blockDim.x + threadIdx.x;
  const int stride = gridDim.x * blockDim.x;
  const int total = E << logF;
  for (; idx < total; idx += stride) {
    int e = idx >> logF;
    int f = idx & (F - 1);
    int s = src[e];
    int d = dst[e];
    float norm = dinv[s] * dinv[d];
    atomicAdd(&out[(size_t)d * F + f], h[(size_t)s * F + f] * norm);
  }
}

// ---- head: y[i] = relu(o2[i,:]) . Wo + bo ----------------------------------
__global__ void k_head(const float* __restrict__ o2, const float* __restrict__ Wo,
                       const float* __restrict__ bo, float* __restrict__ y, int n) {
  int i = blockIdx.x * blockDim.x + threadIdx.x;
  if (i >= n) return;
  const float4* r = (const float4*)(o2 + (size_t)i * 16);
  const float4* w = (const float4*)Wo;
  float s = bo[0];
#pragma unroll
  for (int q = 0; q < 4; ++q) {
    float4 rv = r[q];
    float4 wv = w[q];
    s += fmaxf(rv.x, 0.0f) * wv.x + fmaxf(rv.y, 0.0f) * wv.y +
         fmaxf(rv.z, 0.0f) * wv.z + fmaxf(rv.w, 0.0f) * wv.w;
  }
  y[i] = s;
}

// ---------------------------------------------------------------------------
extern "C" void kernel_launch(void* const* d_in, const int* in_sizes, int n_in,
                              void* d_out, int out_size, void* d_ws, size_t ws_size,
                              hipStream_t stream) {
  const float* x  = (const float*)d_in[0];
  const int*   ei = (const int*)d_in[1];  // edge_index [2, E] int32
  const float* W1 = (const float*)d_in[2];
  const float* b1 = (const float*)d_in[3];
  const float* W2 = (const float*)d_in[4];
  const float* b2 = (const float*)d_in[5];
  const float* Wo = (const float*)d_in[6];
  const float* bo = (const float*)d_in[7];
  float* y = (float*)d_out;

  const int n = in_sizes[0] / 128;
  const int E = in_sizes[1] / 2;
  const int* src = ei;
  const int* dst = ei + E;

  // workspace carve-out (256B aligned): dinv, h1, o1, h2, o2  (~39 MB total)
  char* ws = (char*)d_ws;
  size_t off = 0;
  auto alloc = [&](size_t bytes) -> void* {
    void* p = (void*)(ws + off);
    off += (bytes + 255) & ~(size_t)255;
    return p;
  };
  float* dinv = (float*)alloc((size_t)n * sizeof(float));
  float* h1   = (float*)alloc((size_t)n * 32 * sizeof(float));
  float* o1   = (float*)alloc((size_t)n * 32 * sizeof(float));
  float* h2   = (float*)alloc((size_t)n * 16 * sizeof(float));
  float* o2   = (float*)alloc((size_t)n * 16 * sizeof(float));
  (void)ws_size;
  (void)n_in;
  (void)out_size;

  const int T = 256;
  // normalization (shared by both layers)
  k_fill_one<<<(n + T - 1) / T, T, 0, stream>>>(dinv, n);
  k_deg_accum<<<4096, T, 0, stream>>>(dst, dinv, E);
  k_rsqrt_inplace<<<(n + T - 1) / T, T, 0, stream>>>(dinv, n);

  // layer 1
  {
    int waves = (n + 15) / 16;
    k_gemm1<<<(waves + 7) / 8, T, 0, stream>>>(x, W1, h1, n);
  }
  k_selfinit<<<((n * 8) + T - 1) / T, T, 0, stream>>>((const float4*)h1, dinv,
                                                      (const float4*)b1, (float4*)o1, n, 5);
  k_agg<<<16384, T, 0, stream>>>(src, dst, dinv, h1, o1, E, 5);

  // layer 2 (ReLU of o1 fused into GEMM2 A-load)
  {
    int waves = (n + 15) / 16;
    k_gemm2<<<(waves + 7) / 8, T, 0, stream>>>(o1, W2, h2, n);
  }
  k_selfinit<<<((n * 4) + T - 1) / T, T, 0, stream>>>((const float4*)h2, dinv,
                                                      (const float4*)b2, (float4*)o2, n, 4);
  k_agg<<<16384, T, 0, stream>>>(src, dst, dinv, h2, o2, E, 4);

  // head
  k_head<<<(n + T - 1) / T, T, 0, stream>>>(o2, Wo, bo, y, n);
}